// BitchyNetwork_5239860101610
// MI455X (gfx1250) — compile-verified
//
#include <hip/hip_runtime.h>

#define NN 8192
#define DD 2048
#define FF 16384
#define KK 64

typedef __attribute__((ext_vector_type(2))) float v2f;
typedef __attribute__((ext_vector_type(8))) float v8f;
typedef __attribute__((ext_vector_type(4))) int   v4i;

#if __has_builtin(__builtin_amdgcn_global_load_async_to_lds_b128)
#define HAVE_ASYNC_LDS 1
#else
#define HAVE_ASYNC_LDS 0
#endif
#if __has_builtin(__builtin_amdgcn_s_wait_asynccnt)
#define HAVE_WAIT_ASYNC 1
#else
#define HAVE_WAIT_ASYNC 0
#endif

// builtin signature (from clang-22 diagnostic): param0 = AS(1) int4* (global),
// param1 = LDS side; immediates: offset, cpol
#define GPTR4(p) ((__attribute__((address_space(1))) v4i*)(p))
#define LPTR4(p) ((__attribute__((address_space(3))) v4i*)(p))

// copy 16 bytes global -> LDS (per lane); async on CDNA5, sync fallback otherwise
__device__ __forceinline__ void stage16(float* lds_dst, const float* gsrc) {
#if HAVE_ASYNC_LDS
  __builtin_amdgcn_global_load_async_to_lds_b128(GPTR4(gsrc), LPTR4(lds_dst), 0, 0);
#else
  const float4 v = *(const float4*)gsrc;
  lds_dst[0] = v.x; lds_dst[1] = v.y; lds_dst[2] = v.z; lds_dst[3] = v.w;
#endif
}

template <int N>
__device__ __forceinline__ void wait_async() {
#if HAVE_WAIT_ASYNC
  __builtin_amdgcn_s_wait_asynccnt((unsigned short)N);
#elif HAVE_ASYNC_LDS
  asm volatile("s_wait_asynccnt %0" ::"n"(N) : "memory");
#endif
}

// ---------------------------------------------------------------------------
// Kernel 1: raw = x @ W^T, written into the dense acts region of d_out.
// Block tile 64(n) x 128(f), 256 threads = 8 waves (wave32).
// Wave tile: 16(n) x 64(f) via 4 WMMA f32 16x16x4 accumulators.
// K staged in double-buffered LDS via async global->LDS B128 copies.
// Row pad 36 floats: 16B-aligned rows (async B128), bank-conflict-free frags.
// ---------------------------------------------------------------------------
#define XPAD 36
__global__ void __launch_bounds__(256) sae_gemm_raw(const float* __restrict__ x,
                                                    const float* __restrict__ W,
                                                    float* __restrict__ acts) {
  __shared__ float xs[2][64 * XPAD];
  __shared__ float ws[2][128 * XPAD];

  const int tid  = threadIdx.x;
  const int wave = tid >> 5;
  const int lane = tid & 31;
  const int half = lane >> 4;     // 0: K=0,1  1: K=2,3 (16x4 f32 A/B layout)
  const int l    = lane & 15;
  const int n_off = (wave & 3) * 16;
  const int f_off = (wave >> 2) * 64;
  const int block_n = blockIdx.y * 64;
  const int block_f = blockIdx.x * 128;

  v8f acc[4];
#pragma unroll
  for (int t = 0; t < 4; ++t) {
    v8f z = {0.f, 0.f, 0.f, 0.f, 0.f, 0.f, 0.f, 0.f};
    acc[t] = z;
  }

  // 6 stage16 instructions per wave per chunk: 2 for x (64x32), 4 for W (128x32)
  auto prefetch = [&](int buf, int d0) {
#pragma unroll
    for (int p = 0; p < 2; ++p) {
      const int idx = tid + p * 256;
      const int row = idx >> 3;
      const int c4  = (idx & 7) * 4;
      stage16(&xs[buf][row * XPAD + c4],
              x + (size_t)(block_n + row) * DD + d0 + c4);
    }
#pragma unroll
    for (int p = 0; p < 4; ++p) {
      const int idx = tid + p * 256;
      const int row = idx >> 3;
      const int c4  = (idx & 7) * 4;
      stage16(&ws[buf][row * XPAD + c4],
              W + (size_t)(block_f + row) * DD + d0 + c4);
    }
  };

  constexpr int CHUNKS = DD / 32;
  prefetch(0, 0);
  int buf = 0;
  for (int c = 0; c < CHUNKS; ++c) {
    if (c + 1 < CHUNKS) {
      prefetch(buf ^ 1, (c + 1) * 32);   // next chunk in flight
      wait_async<6>();                   // drain current chunk's 6 copies
    } else {
      wait_async<0>();
    }
    __syncthreads();                     // current chunk visible to all waves

#pragma unroll
    for (int kk = 0; kk < 32; kk += 4) {
      const v2f a = *(const v2f*)&xs[buf][(n_off + l) * XPAD + kk + 2 * half];
#pragma unroll
      for (int t = 0; t < 4; ++t) {
        const v2f b = *(const v2f*)&ws[buf][(f_off + t * 16 + l) * XPAD + kk + 2 * half];
        acc[t] = __builtin_amdgcn_wmma_f32_16x16x4_f32(
            /*neg_a=*/false, a, /*neg_b=*/false, b,
            /*c_mod=*/(short)0, acc[t], /*reuse_a=*/false, /*reuse_b=*/false);
      }
    }
    __syncthreads();                     // WAR: done reading before next overwrite
    buf ^= 1;
  }

  // C/D layout: VGPR r -> M=r (lanes 0-15) / M=r+8 (lanes 16-31), N=l
#pragma unroll
  for (int t = 0; t < 4; ++t) {
    const int f = block_f + f_off + t * 16 + l;
#pragma unroll
    for (int r = 0; r < 8; ++r) {
      const int n = block_n + n_off + r + 8 * half;
      acts[(size_t)n * FF + f] = acc[t][r];   // b32 stores (acts base is 4B aligned)
    }
  }
}

// ---------------------------------------------------------------------------
// Kernel 2: per-row exact top-64 via 4-pass MSB radix select on order-
// preserving uint keys. Zeros losers in acts in place; exact tie-break by
// index; emits compact (idx, val) winner lists to workspace.
// ---------------------------------------------------------------------------
__global__ void __launch_bounds__(256) sae_topk(float* __restrict__ acts,
                                                int* __restrict__ win_idx,
                                                float* __restrict__ win_val) {
  const int n   = blockIdx.x;
  const int tid = threadIdx.x;
  float* row = acts + (size_t)n * FF;

  __shared__ unsigned hist[256];
  __shared__ int scan[256];
  __shared__ unsigned sh_pfx;
  __shared__ int sh_rem, sh_gt, sh_slot;

  // each thread owns a contiguous chunk of 64 indices (keeps tie order simple)
  unsigned key[64];
#pragma unroll
  for (int j = 0; j < 64; ++j) {
    const unsigned b = __float_as_uint(row[tid * 64 + j]);
    key[j] = b ^ ((b & 0x80000000u) ? 0xFFFFFFFFu : 0x80000000u);
  }

  if (tid == 0) { sh_pfx = 0u; sh_rem = KK; sh_gt = 0; sh_slot = 0; }
  __syncthreads();

  for (int pass = 0; pass < 4; ++pass) {
    const int shift = 24 - pass * 8;
    hist[tid] = 0u;
    __syncthreads();
    const unsigned pmask = (pass == 0) ? 0u : (0xFFFFFFFFu << (shift + 8));
    const unsigned pfx = sh_pfx;
#pragma unroll
    for (int j = 0; j < 64; ++j) {
      if ((key[j] & pmask) == pfx)
        atomicAdd(&hist[(key[j] >> shift) & 255u], 1u);
    }
    __syncthreads();
    if (tid == 0) {
      const int rem = sh_rem;
      int above = 0;
      int b = 255;
      for (; b > 0; --b) {
        const int c = (int)hist[b];
        if (above + c >= rem) break;
        above += c;
      }
      sh_pfx = pfx | ((unsigned)b << shift);
      sh_rem = rem - above;
      sh_gt += above;
    }
    __syncthreads();
  }

  const unsigned thresh = sh_pfx;        // exact key of the 64th largest
  const int ties_needed = KK - sh_gt;    // equals (by index order) to keep

  int local_eq = 0;
#pragma unroll
  for (int j = 0; j < 64; ++j) local_eq += (key[j] == thresh) ? 1 : 0;
  scan[tid] = local_eq;
  __syncthreads();
  if (tid == 0) {                        // serial exclusive scan (256 entries)
    int s = 0;
    for (int t = 0; t < 256; ++t) { const int c = scan[t]; scan[t] = s; s += c; }
  }
  __syncthreads();
  int eqb = scan[tid];

#pragma unroll
  for (int j = 0; j < 64; ++j) {
    const int i = tid * 64 + j;
    const unsigned u = key[j];
    bool win = false;
    if (u > thresh) {
      win = true;
    } else if (u == thresh) {
      if (eqb < ties_needed) win = true;
      eqb++;
    }
    if (win) {
      const int s = atomicAdd(&sh_slot, 1);
      const unsigned b = (u & 0x80000000u) ? (u ^ 0x80000000u) : ~u; // invert key
      win_idx[n * KK + s] = i;
      win_val[n * KK + s] = __uint_as_float(b);
    } else {
      row[i] = 0.0f;
    }
  }
}

// ---------------------------------------------------------------------------
// Kernel 3: recon = acts_sparse @ W (64 gathered W rows per token, L2 resident)
// plus per-row squared error reduced into loss.
// ---------------------------------------------------------------------------
__global__ void __launch_bounds__(256) sae_recon_loss(const float* __restrict__ x,
                                                      const float* __restrict__ W,
                                                      const int* __restrict__ win_idx,
                                                      const float* __restrict__ win_val,
                                                      float* __restrict__ recon,
                                                      float* __restrict__ loss) {
  const int n   = blockIdx.x;
  const int tid = threadIdx.x;
  __shared__ int   sidx[KK];
  __shared__ float sval[KK];
  __shared__ float red[256];

  if (tid < KK) {
    sidx[tid] = win_idx[n * KK + tid];
    sval[tid] = win_val[n * KK + tid];
  }
  __syncthreads();

  float acc[8];
#pragma unroll
  for (int j = 0; j < 8; ++j) acc[j] = 0.0f;

  for (int t = 0; t < KK; ++t) {
    const float* wr = W + (size_t)sidx[t] * DD;
    const float v = sval[t];
#pragma unroll
    for (int j = 0; j < 8; ++j) acc[j] += v * wr[tid + j * 256];
  }

  float lsum = 0.0f;
#pragma unroll
  for (int j = 0; j < 8; ++j) {
    const int d = tid + j * 256;
    const float diff = acc[j] - x[(size_t)n * DD + d];
    lsum += diff * diff;
    recon[(size_t)n * DD + d] = acc[j];   // recon base is +1 float: b32 stores
  }

  red[tid] = lsum;
  __syncthreads();
  for (int s = 128; s > 0; s >>= 1) {
    if (tid < s) red[tid] += red[tid + s];
    __syncthreads();
  }
  if (tid == 0) atomicAdd(loss, red[0] * (1.0f / (float)NN));
}

// ---------------------------------------------------------------------------
extern "C" void kernel_launch(void* const* d_in, const int* in_sizes, int n_in,
                              void* d_out, int out_size, void* d_ws, size_t ws_size,
                              hipStream_t stream) {
  const float* x = (const float*)d_in[0];   // [N, D]
  const float* W = (const float*)d_in[1];   // [F, D]

  float* out   = (float*)d_out;
  float* loss  = out;                               // 1 element
  float* recon = out + 1;                           // [N, D]
  float* acts  = out + 1 + (size_t)NN * DD;         // [N, F]

  int*   win_idx = (int*)d_ws;                                            // [N, K]
  float* win_val = (float*)((char*)d_ws + (size_t)NN * KK * sizeof(int)); // [N, K]

  (void)hipMemsetAsync(loss, 0, sizeof(float), stream);

  dim3 gemm_grid(FF / 128, NN / 64);
  sae_gemm_raw<<<gemm_grid, 256, 0, stream>>>(x, W, acts);
  sae_topk<<<NN, 256, 0, stream>>>(acts, win_idx, win_val);
  sae_recon_loss<<<NN, 256, 0, stream>>>(x, W, win_idx, win_val, recon, loss);
}